// GATModel_57818849738884
// MI455X (gfx1250) — compile-verified
//
#include <hip/hip_runtime.h>

// ---------------------------------------------------------------------------
// GAT (2-layer) for MI455X / gfx1250.
//  - Dense GEMMs -> V_WMMA_F32_16X16X4_F32 (CDNA5 fp32 WMMA shape)
//  - Weights transposed once in global, then staged into LDS by the Tensor
//    Data Mover with pad_enable (stride 256 -> 260 dwords): bank-conflict-free
//    ds_load_b64 B operands.
//  - GEMM inner loop is software-pipelined (ping-pong operand sets) with
//    __builtin_amdgcn_sched_barrier fences so ds latency overlaps WMMA issue.
//  - Edge passes: float4 (b128) gathers, u32-monotonic atomicMax segment-max,
//    fused exp/denom/weighted-V scatter; per-node normalize afterwards.
// ---------------------------------------------------------------------------

typedef __attribute__((ext_vector_type(2))) float v2f;
typedef __attribute__((ext_vector_type(8))) float v8f;
typedef __attribute__((ext_vector_type(4))) unsigned int v4u;
typedef __attribute__((ext_vector_type(4))) int v4i;
typedef __attribute__((ext_vector_type(8))) int v8i;

#define WT_STRIDE 260  // 256 dwords + 4 dwords TDM padding per row in LDS
#define WT_LDS_DW (64 * WT_STRIDE)  // dwords per staged matrix (66560 B)

__device__ __forceinline__ v8f wmma_f32_16x16x4(v2f a, v2f b, v8f c) {
  // 8 args: (neg_a, A, neg_b, B, c_mod, C, reuse_a, reuse_b)
  return __builtin_amdgcn_wmma_f32_16x16x4_f32(false, a, false, b, (short)0, c,
                                               false, false);
}

// Monotonic float <-> uint mapping so unsigned atomicMax implements float max.
__device__ __forceinline__ unsigned f2u_mono(float f) {
  unsigned u = __float_as_uint(f);
  return (u & 0x80000000u) ? ~u : (u | 0x80000000u);
}
__device__ __forceinline__ float u2f_mono(unsigned u) {
  return (u & 0x80000000u) ? __uint_as_float(u & 0x7FFFFFFFu)
                           : __uint_as_float(~u);
}

// ---------------------------------------------------------------------------
// TDM: DMA a [64 x 256] f32 row-major matrix (transposed weights) from global
// into LDS at byte offset lds_off, inserting 4 dwords of padding every 256
// dwords (pad_interval=7 -> 256 dwords, pad_amount=3 -> 4 dwords) so the LDS
// row stride is 260 dwords. D# bitfields per CDNA5 ISA.
// ---------------------------------------------------------------------------
__device__ __forceinline__ void tdm_load_wt(const float* gsrc,
                                            unsigned lds_off) {
  unsigned long long ga = (unsigned long long)(size_t)gsrc;
  v4u g0;
  g0.x = 1u;                                                 // count=1
  g0.y = lds_off;                                            // lds_addr
  g0.z = (unsigned)(ga & 0xffffffffu);                       // global_addr lo
  g0.w = (unsigned)((ga >> 32) & 0x01ffffffu) | (2u << 30);  // addr hi|type=2
  v8i g1;
  g1[0] = (int)((2u << 16) | (1u << 20) | (7u << 22) | (3u << 25));
  g1[1] = (int)(256u << 16);  // tensor_dim0 = 256 (bits 63:48)
  g1[2] = (int)(64u << 16);   // dim0 hi = 0 | tensor_dim1 lo = 64
  g1[3] = (int)(256u << 16);  // dim1 hi = 0 | tile_dim0 = 256
  g1[4] = (int)64;            // tile_dim1 = 64 | tile_dim2 = 0
  g1[5] = 256;                // tensor_dim0_stride = 256 (lo32)
  g1[6] = 0;
  g1[7] = 0;
  v4i gz4 = {0, 0, 0, 0};
  v8i gz8 = {0, 0, 0, 0, 0, 0, 0, 0};
  __builtin_amdgcn_tensor_load_to_lds(g0, g1, gz4, gz4, gz8, 0);
}

// One-shot weight transpose: W [256,64] -> Wt [64,256].
__global__ __launch_bounds__(256) void transpose_w_kernel(
    const float* __restrict__ W, float* __restrict__ Wt) {
  const int t = blockIdx.x * blockDim.x + threadIdx.x;
  if (t >= 256 * 64) return;
  const int k = t >> 6;
  const int c = t & 63;
  Wt[c * 256 + k] = W[k * 64 + c];
}

// ---------------------------------------------------------------------------
// Layer-0 fused GEMM: Q = relu(X*Wq + bq), K = relu(X*Wk + bk), V = X*Wv.
// X: [N,256], transposed weights Wt*: [64,256]. Wave handles a 16-row tile.
// Software-pipelined: while one operand set feeds 12 WMMAs, the other set's
// 12 ds_load_b64 (+1 global_load_b64) are in flight. sched_barrier(0) fences
// keep the scheduler from re-serializing load->use pairs.
// ---------------------------------------------------------------------------
__global__ __launch_bounds__(256) void gemm_qkv0_kernel(
    const float* __restrict__ X,
    const float* __restrict__ WTq, const float* __restrict__ bq,
    const float* __restrict__ WTk, const float* __restrict__ bk,
    const float* __restrict__ WTv,
    float* __restrict__ Q, float* __restrict__ K, float* __restrict__ V,
    int N) {
  extern __shared__ float smem[];  // 3 * 64*260 dwords = 199,680 B
  const unsigned lds_base = (unsigned)(unsigned long long)(size_t)&smem[0];
  const int lane = threadIdx.x & 31;
  const int wave = threadIdx.x >> 5;

  if (wave == 0 && lane == 0) {  // TDM ignores EXEC; issue once per workgroup
    tdm_load_wt(WTq, lds_base + 0u);
    tdm_load_wt(WTk, lds_base + WT_LDS_DW * 4u);
    tdm_load_wt(WTv, lds_base + 2u * WT_LDS_DW * 4u);
    __builtin_amdgcn_s_wait_tensorcnt(0);
  }
  __syncthreads();

  const int row0 = (blockIdx.x * 8 + wave) * 16;
  if (row0 >= N) return;
  const int m = lane & 15;
  const int half = lane >> 4;
  int arow = row0 + m;
  if (arow >= N) arow = N - 1;  // clamp: EXEC must stay all-ones for WMMA
  const float* __restrict__ xrow = X + (size_t)arow * 256;
  const float* __restrict__ wqL = smem;
  const float* __restrict__ wkL = smem + WT_LDS_DW;
  const float* __restrict__ wvL = smem + 2 * WT_LDS_DW;

  v8f aq[4] = {}, ak[4] = {}, av[4] = {};

  auto loadB = [&](int k, v2f(&q_)[4], v2f(&k_)[4], v2f(&v_)[4]) {
    const int ka = k + half * 2;
#pragma unroll
    for (int nt = 0; nt < 4; ++nt) {
      const int wo = (nt * 16 + m) * WT_STRIDE + ka;  // -> ds_load_b64
      q_[nt] = *(const v2f*)(wqL + wo);
      k_[nt] = *(const v2f*)(wkL + wo);
      v_[nt] = *(const v2f*)(wvL + wo);
    }
  };
  auto step = [&](v2f a, const v2f(&q_)[4], const v2f(&k_)[4],
                  const v2f(&v_)[4]) {
#pragma unroll
    for (int nt = 0; nt < 4; ++nt) {
      aq[nt] = wmma_f32_16x16x4(a, q_[nt], aq[nt]);
      ak[nt] = wmma_f32_16x16x4(a, k_[nt], ak[nt]);
      av[nt] = wmma_f32_16x16x4(a, v_[nt], av[nt]);
    }
  };

  v2f AQ[4], AK[4], AV[4], BQ[4], BK[4], BV[4];
  v2f aA = *(const v2f*)(xrow + half * 2);
  loadB(0, AQ, AK, AV);
  for (int k = 0; k < 256; k += 8) {
    // Preload operands for step k+4 into the pong set.
    const v2f aB = *(const v2f*)(xrow + (k + 4) + half * 2);
    loadB(k + 4, BQ, BK, BV);
    __builtin_amdgcn_sched_barrier(0);
    step(aA, AQ, AK, AV);  // waits only on ping set (already returned)
    __builtin_amdgcn_sched_barrier(0);
    // Preload operands for step k+8 into the ping set (wraps on last iter).
    const int k2 = (k + 8) & 255;
    aA = *(const v2f*)(xrow + k2 + half * 2);
    loadB(k2, AQ, AK, AV);
    __builtin_amdgcn_sched_barrier(0);
    step(aB, BQ, BK, BV);
    __builtin_amdgcn_sched_barrier(0);
  }

#pragma unroll
  for (int nt = 0; nt < 4; ++nt) {
    const int col = nt * 16 + m;
    const float biasq = bq[col];
    const float biask = bk[col];
#pragma unroll
    for (int r = 0; r < 8; ++r) {
      const int row = row0 + r + 8 * half;
      if (row < N) {
        Q[(size_t)row * 64 + col] = fmaxf(aq[nt][r] + biasq, 0.f);
        K[(size_t)row * 64 + col] = fmaxf(ak[nt][r] + biask, 0.f);
        V[(size_t)row * 64 + col] = av[nt][r];
      }
    }
  }
}

// Layer-1 V GEMM: V1 = H*Wv1, H: [N,64], Wv1: [64,16]. One col tile, 16 k-steps.
__global__ __launch_bounds__(256) void gemm_v1_kernel(
    const float* __restrict__ Hm, const float* __restrict__ Wv,
    float* __restrict__ V, int N) {
  const int lane = threadIdx.x & 31;
  const int wave = threadIdx.x >> 5;
  const int row0 = (blockIdx.x * 8 + wave) * 16;
  if (row0 >= N) return;
  const int m = lane & 15;
  const int half = lane >> 4;
  int arow = row0 + m;
  if (arow >= N) arow = N - 1;
  const float* __restrict__ hrow = Hm + (size_t)arow * 64;
  v8f acc = {};
#pragma unroll
  for (int k = 0; k < 64; k += 4) {
    const int ka = k + half * 2;
    const v2f a = *(const v2f*)(hrow + ka);
    v2f b;
    b.x = Wv[(size_t)ka * 16 + m];
    b.y = Wv[(size_t)(ka + 1) * 16 + m];
    acc = wmma_f32_16x16x4(a, b, acc);
  }
#pragma unroll
  for (int r = 0; r < 8; ++r) {
    const int row = row0 + r + 8 * half;
    if (row < N) V[(size_t)row * 16 + m] = acc[r];
  }
}

// Layer-1 attention GEMVs: Q1 = relu(H*Wq1 + bq1), K1 = relu(H*Wk1 + bk1).
__global__ __launch_bounds__(256) void qk1_kernel(
    const float* __restrict__ Hm, const float* __restrict__ Wq,
    const float* __restrict__ bq, const float* __restrict__ Wk,
    const float* __restrict__ bk, float* __restrict__ Q1,
    float* __restrict__ K1, int N) {
  const int n = blockIdx.x * blockDim.x + threadIdx.x;
  if (n >= N) return;
  const float4* __restrict__ hr = (const float4*)(Hm + (size_t)n * 64);
  const float4* __restrict__ wq4 = (const float4*)Wq;
  const float4* __restrict__ wk4 = (const float4*)Wk;
  float q = 0.f, k = 0.f;
#pragma unroll
  for (int j = 0; j < 16; ++j) {
    const float4 x = hr[j];
    const float4 a = wq4[j];
    const float4 b = wk4[j];
    q += x.x * a.x + x.y * a.y + x.z * a.z + x.w * a.w;
    k += x.x * b.x + x.y * b.y + x.z * b.z + x.w * b.w;
  }
  Q1[n] = fmaxf(q + bq[0], 0.f);
  K1[n] = fmaxf(k + bk[0], 0.f);
}

__device__ __forceinline__ float dot4(float4 a, float4 b) {
  return a.x * b.x + a.y * b.y + a.z * b.z + a.w * b.w;
}

// ---------------------------------------------------------------------------
// Edge pass 1: score + segment max. Thread t = (edge, head); rows are
// 32B-aligned so Q/K gathers are b128 loads.
// ---------------------------------------------------------------------------
template <int H, int DA>
__global__ __launch_bounds__(256) void edge_max_kernel(
    const int* __restrict__ ei, int E, int N, const float* __restrict__ Q,
    const float* __restrict__ K, unsigned* __restrict__ smax) {
  const long long t = (long long)blockIdx.x * blockDim.x + threadIdx.x;
  const long long tot = (long long)(E + N) * H;
  if (t >= tot) return;
  const int h = (int)(t % H);
  const long long e = t / H;
  int r, c;
  if (e < E) {
    r = ei[e];
    c = ei[(size_t)E + e];
  } else {
    r = c = (int)(e - E);  // self-loop
  }
  float s = 0.f;
  if (DA == 8) {
    const float4* __restrict__ q4 = (const float4*)(Q + (size_t)r * 64 + h * 8);
    const float4* __restrict__ k4 = (const float4*)(K + (size_t)c * 64 + h * 8);
    s = dot4(q4[0], k4[0]) + dot4(q4[1], k4[1]);
  } else {
    s = Q[r] * K[c];
  }
  atomicMax(&smax[(size_t)r * H + h], f2u_mono(s));
}

// ---------------------------------------------------------------------------
// Edge pass 2 (fused): recompute score, ex = exp(score - max), accumulate
// denom[row,h] += ex  and  msg[row,h,:] += ex * V[col,h,:]  (unnormalized).
// ---------------------------------------------------------------------------
template <int H, int DA, int DV>
__global__ __launch_bounds__(256) void edge_accum_kernel(
    const int* __restrict__ ei, int E, int N, const float* __restrict__ Q,
    const float* __restrict__ K, const float* __restrict__ V,
    const unsigned* __restrict__ smax, float* __restrict__ denom,
    float* __restrict__ msg) {
  const long long t = (long long)blockIdx.x * blockDim.x + threadIdx.x;
  const long long tot = (long long)(E + N) * H;
  if (t >= tot) return;
  const int h = (int)(t % H);
  const long long e = t / H;
  int r, c;
  if (e < E) {
    r = ei[e];
    c = ei[(size_t)E + e];
  } else {
    r = c = (int)(e - E);
  }
  float s = 0.f;
  if (DA == 8) {
    const float4* __restrict__ q4 = (const float4*)(Q + (size_t)r * 64 + h * 8);
    const float4* __restrict__ k4 = (const float4*)(K + (size_t)c * 64 + h * 8);
    s = dot4(q4[0], k4[0]) + dot4(q4[1], k4[1]);
  } else {
    s = Q[r] * K[c];
  }
  const float mx = u2f_mono(smax[(size_t)r * H + h]);
  const float ex = __expf(s - mx);
  atomicAdd(&denom[(size_t)r * H + h], ex);
  const float4* __restrict__ v4 =
      (const float4*)(V + (size_t)c * (H * DV) + h * DV);
  float* __restrict__ mrow = msg + (size_t)r * (H * DV) + h * DV;
#pragma unroll
  for (int d4 = 0; d4 < DV / 4; ++d4) {
    const float4 v = v4[d4];
    atomicAdd(&mrow[d4 * 4 + 0], ex * v.x);
    atomicAdd(&mrow[d4 * 4 + 1], ex * v.y);
    atomicAdd(&mrow[d4 * 4 + 2], ex * v.z);
    atomicAdd(&mrow[d4 * 4 + 3], ex * v.w);
  }
}

// Per-node normalize + bias (+ optional relu): out = [relu](msg/denom + b).
template <int H, int DV>
__global__ __launch_bounds__(256) void finalize_kernel(
    const float* __restrict__ msg, const float* __restrict__ denom,
    const float* __restrict__ bias, float* __restrict__ out, int N,
    bool doRelu) {
  const long long t = (long long)blockIdx.x * blockDim.x + threadIdx.x;
  const long long tot = (long long)N * H * DV;
  if (t >= tot) return;
  const int n = (int)(t / (H * DV));
  const int c = (int)(t % (H * DV));
  const int h = c / DV;
  float v = msg[t] / denom[(size_t)n * H + h] + bias[c];
  if (doRelu) v = fmaxf(v, 0.f);
  out[t] = v;
}

__global__ __launch_bounds__(256) void zero_kernel(unsigned* __restrict__ p,
                                                   long long n) {
  const long long t = (long long)blockIdx.x * blockDim.x + threadIdx.x;
  if (t < n) p[t] = 0u;
}

static inline int nblk(long long n, int tpb) {
  return (int)((n + tpb - 1) / tpb);
}

extern "C" void kernel_launch(void* const* d_in, const int* in_sizes, int n_in,
                              void* d_out, int out_size, void* d_ws,
                              size_t ws_size, hipStream_t stream) {
  (void)n_in;
  (void)out_size;
  (void)ws_size;
  const float* x = (const float*)d_in[0];
  const int* ei = (const int*)d_in[1];
  const float* Wq0 = (const float*)d_in[2];
  const float* bq0 = (const float*)d_in[3];
  const float* Wk0 = (const float*)d_in[4];
  const float* bk0 = (const float*)d_in[5];
  const float* Wv0 = (const float*)d_in[6];
  const float* b0 = (const float*)d_in[7];
  const float* Wq1 = (const float*)d_in[8];
  const float* bq1 = (const float*)d_in[9];
  const float* Wk1 = (const float*)d_in[10];
  const float* bk1 = (const float*)d_in[11];
  const float* Wv1 = (const float*)d_in[12];
  const float* b1 = (const float*)d_in[13];

  const int F = 256;
  const int N = in_sizes[0] / F;
  const int E = in_sizes[1] / 2;
  const int TPB = 256;

  // Workspace carve-up (256B-aligned chunks).
  size_t off = 0;
  float* base = (float*)d_ws;
  auto alloc = [&](size_t nfloats) -> float* {
    float* p = base + off;
    off += (nfloats + 63) & ~(size_t)63;
    return p;
  };
  float* Q0 = alloc((size_t)N * 64);
  float* K0 = alloc((size_t)N * 64);
  float* V0 = alloc((size_t)N * 64);
  float* H1 = alloc((size_t)N * 64);
  float* MSG = alloc((size_t)N * 64);
  unsigned* SMAX = (unsigned*)alloc((size_t)N * 8);
  float* DEN = alloc((size_t)N * 8);
  float* Q1 = alloc((size_t)N);
  float* K1 = alloc((size_t)N);
  float* V1 = alloc((size_t)N * 16);
  float* WTq = alloc(256 * 64);  // transposed weights [64,256]
  float* WTk = alloc(256 * 64);
  float* WTv = alloc(256 * 64);

  const int tiles = (N + 15) / 16;
  const int gemmBlocks = (tiles + 7) / 8;
  const long long tot0 = (long long)(E + N) * 8;
  const long long tot1 = (long long)(E + N);
  const size_t ldsBytes = 3u * WT_LDS_DW * sizeof(float);  // 199,680 B

  // ---------------- Layer 0 (8 heads, att dim 8, value dim 8) -------------
  zero_kernel<<<nblk((long long)N * 8, TPB), TPB, 0, stream>>>(
      SMAX, (long long)N * 8);
  zero_kernel<<<nblk((long long)N * 8, TPB), TPB, 0, stream>>>(
      (unsigned*)DEN, (long long)N * 8);
  zero_kernel<<<nblk((long long)N * 64, TPB), TPB, 0, stream>>>(
      (unsigned*)MSG, (long long)N * 64);
  transpose_w_kernel<<<nblk(256 * 64, TPB), TPB, 0, stream>>>(Wq0, WTq);
  transpose_w_kernel<<<nblk(256 * 64, TPB), TPB, 0, stream>>>(Wk0, WTk);
  transpose_w_kernel<<<nblk(256 * 64, TPB), TPB, 0, stream>>>(Wv0, WTv);
  gemm_qkv0_kernel<<<gemmBlocks, 256, ldsBytes, stream>>>(
      x, WTq, bq0, WTk, bk0, WTv, Q0, K0, V0, N);
  edge_max_kernel<8, 8><<<nblk(tot0, TPB), TPB, 0, stream>>>(ei, E, N, Q0, K0,
                                                             SMAX);
  edge_accum_kernel<8, 8, 8><<<nblk(tot0, TPB), TPB, 0, stream>>>(
      ei, E, N, Q0, K0, V0, SMAX, DEN, MSG);
  finalize_kernel<8, 8><<<nblk((long long)N * 64, TPB), TPB, 0, stream>>>(
      MSG, DEN, b0, H1, N, true);

  // ---------------- Layer 1 (1 head, att dim 1, value dim 16) -------------
  zero_kernel<<<nblk((long long)N, TPB), TPB, 0, stream>>>(SMAX, (long long)N);
  zero_kernel<<<nblk((long long)N, TPB), TPB, 0, stream>>>((unsigned*)DEN,
                                                           (long long)N);
  zero_kernel<<<nblk((long long)N * 16, TPB), TPB, 0, stream>>>(
      (unsigned*)MSG, (long long)N * 16);
  qk1_kernel<<<nblk((long long)N, TPB), TPB, 0, stream>>>(H1, Wq1, bq1, Wk1,
                                                          bk1, Q1, K1, N);
  gemm_v1_kernel<<<gemmBlocks, 256, 0, stream>>>(H1, Wv1, V1, N);
  edge_max_kernel<1, 1><<<nblk(tot1, TPB), TPB, 0, stream>>>(ei, E, N, Q1, K1,
                                                             SMAX);
  edge_accum_kernel<1, 1, 16><<<nblk(tot1, TPB), TPB, 0, stream>>>(
      ei, E, N, Q1, K1, V1, SMAX, DEN, MSG);
  finalize_kernel<1, 16><<<nblk((long long)N * 16, TPB), TPB, 0, stream>>>(
      MSG, DEN, b1, (float*)d_out, N, false);
}